// DSQGAttention_9543417331817
// MI455X (gfx1250) — compile-verified
//
#include <hip/hip_runtime.h>
#include <hip/hip_bf16.h>

// ---------------------------------------------------------------------------
// DSQG attention forward for MI455X (gfx1250, wave32, WMMA).
//   B=2, T=2048, C=768, H=12, HD=64, J=52 offsets.
//   Pipeline:  x @ Wqkv^T  ->  gather/softmax attention  ->  attn @ Wout^T
//   GEMMs: V_WMMA_F32_16X16X4_F32, register double-buffered (K macro-step 16).
//   Attention stages scale_embed with GLOBAL_LOAD_ASYNC_TO_LDS_B128 (ASYNCcnt).
// ---------------------------------------------------------------------------

typedef __attribute__((ext_vector_type(2))) float v2f;
typedef __attribute__((ext_vector_type(4))) float v4f;
typedef __attribute__((ext_vector_type(8))) float v8f;

#define B_   2
#define T_   2048
#define C_   768
#define H_   12
#define HD_  64
#define J_   52
#define SE_PITCH 68   // 272B rows: 16B-aligned, avoids 64-bank conflicts

__constant__ int c_offs[J_] = {
   0, 1, 2, 3, 4, 5, 6, 7, 8, 9,10,11,12,13,14,15,16,17,18,19,
  20,21,22,23,24,25,26,27,28,29,30,31,32,33,34,35,36,37,38,39,
  40, 96,128,145,163,185,209,236,266,301,340,384 };

// ---------------------------------------------------------------------------
// Out(M,N) = A(M,K) @ W(N,K)^T, fp32 WMMA 16x16x4, one wave per 16x64 tile.
// A-frag (ISA 7.12.2): lane<16 -> K={k0,k0+1}, lane>=16 -> K={k0+2,k0+3},
// row = m0+(lane&15). B mirrors A with N=lane&15. C/D: VGPR r holds
// M = r + 8*(lane>=16), N = lane&15.
// Double-buffered over K macro-steps of 16 (4 WMMA-steps each).
// ---------------------------------------------------------------------------
__global__ __launch_bounds__(256) void wmma_gemm_nt(
    const float* __restrict__ A,   // M x K row-major
    const float* __restrict__ W,   // N x K row-major
    float* __restrict__ Out,       // M x N row-major
    int M, int N, int K, int nT64)
{
  const int wave = threadIdx.x >> 5;
  const int lane = threadIdx.x & 31;
  const int task = blockIdx.x * 8 + wave;
  const int mt = task / nT64;
  const int nt = task - mt * nT64;
  const int m0 = mt << 4;
  const int n0 = nt << 6;
  if (m0 >= M) return;

  const int lh = lane & 15;
  const int kh = (lane >> 4) << 1;            // 0 or 2
  const float* ar = A + (size_t)(m0 + lh) * K + kh;
  const float* wr = W + (size_t)(n0 + lh) * K + kh;
  const size_t w16 = (size_t)16 * K;

  v8f acc0 = {}; v8f acc1 = {}; v8f acc2 = {}; v8f acc3 = {};

  v2f aA[4], aB[4];          // A fragments for 4 consecutive K-steps
  v2f bA[4][4], bB[4][4];    // [n-tile][K-step]

  auto load_buf = [&](v2f (&a)[4], v2f (&b)[4][4], int k0) {
    #pragma unroll
    for (int i = 0; i < 4; ++i)
      a[i] = *(const v2f*)(ar + k0 + 4 * i);
    #pragma unroll
    for (int n = 0; n < 4; ++n)
      #pragma unroll
      for (int i = 0; i < 4; ++i)
        b[n][i] = *(const v2f*)(wr + (size_t)n * w16 + k0 + 4 * i);
  };

  auto mma_buf = [&](v2f (&a)[4], v2f (&b)[4][4]) {
    #pragma unroll
    for (int i = 0; i < 4; ++i) {
      acc0 = __builtin_amdgcn_wmma_f32_16x16x4_f32(false, a[i], false, b[0][i], (short)0, acc0, false, false);
      acc1 = __builtin_amdgcn_wmma_f32_16x16x4_f32(false, a[i], false, b[1][i], (short)0, acc1, false, false);
      acc2 = __builtin_amdgcn_wmma_f32_16x16x4_f32(false, a[i], false, b[2][i], (short)0, acc2, false, false);
      acc3 = __builtin_amdgcn_wmma_f32_16x16x4_f32(false, a[i], false, b[3][i], (short)0, acc3, false, false);
    }
  };

  load_buf(aA, bA, 0);
  int k0 = 0;
  for (; k0 + 32 < K; k0 += 32) {
    if (k0 + 144 <= K) {  // prefetch 128 floats ahead, stays inside the rows
      __builtin_prefetch(ar + k0 + 128, 0, 0);
      __builtin_prefetch(wr + k0 + 128, 0, 0);
      __builtin_prefetch(wr + w16 + k0 + 128, 0, 0);
      __builtin_prefetch(wr + 2 * w16 + k0 + 128, 0, 0);
      __builtin_prefetch(wr + 3 * w16 + k0 + 128, 0, 0);
    }
    load_buf(aB, bB, k0 + 16);   // prefetch next macro-step into regs
    mma_buf(aA, bA);             // consume current
    load_buf(aA, bA, k0 + 32);
    mma_buf(aB, bB);
  }
  load_buf(aB, bB, K - 16);      // k0 == K-32 here
  mma_buf(aA, bA);
  mma_buf(aB, bB);

  const int r0 = m0 + ((lane >> 4) << 3);
  float* o = Out + (size_t)r0 * N + n0 + lh;
  #pragma unroll
  for (int r = 0; r < 8; ++r) {
    o[(size_t)r * N +  0] = acc0[r];
    o[(size_t)r * N + 16] = acc1[r];
    o[(size_t)r * N + 32] = acc2[r];
    o[(size_t)r * N + 48] = acc3[r];
  }
}

// ---------------------------------------------------------------------------
// Attention: one wave32 per (b,h,t).
// scale_embed staged via GLOBAL_LOAD_ASYNC_TO_LDS_B128 (ASYNCcnt, no VGPR
// round-trip). Phase 1: lane l scores offsets j=l, j=l+32 with float4 reads.
// Phase 2: shuffle softmax, p to LDS, lane l accumulates dims l and l+32.
// k/v gathers are L2-resident (qkv = 37.7MB << 192MB L2, ~52x row reuse).
// ---------------------------------------------------------------------------
__global__ __launch_bounds__(256) void dsqg_attn(
    const float* __restrict__ qkv,         // (B*T, 3C)
    const float* __restrict__ pos_bias,    // (J, H)
    const float* __restrict__ scale_embed, // (J, HD)
    const float* __restrict__ if_gain,     // (H)
    float* __restrict__ attn)              // (B*T, C)
{
  __shared__ float s_se[J_ * SE_PITCH];
  __shared__ float s_q[8][HD_];
  __shared__ float s_p[8][64];

  // Async DMA scale_embed (52x64 f32) into pitched LDS: 832 b128 chunks.
  // Each float4 chunk stays inside one row (64 % 4 == 0); dest 16B-aligned
  // (pitch 272B). GV addressing: LDS offset in one VGPR, global address in a VGPR pair.
  {
    const int nVec = (J_ * HD_) / 4;                    // 832
    for (int i4 = threadIdx.x; i4 < nVec; i4 += 256) {
      const int j = (i4 * 4) >> 6;
      const int d = (i4 * 4) & 63;
      __attribute__((address_space(3))) float* dst3 =
          (__attribute__((address_space(3))) float*)&s_se[j * SE_PITCH + d];
      unsigned lds_off = (unsigned)(unsigned long long)dst3;
      unsigned long long gaddr =
          (unsigned long long)(const void*)(scale_embed + (size_t)i4 * 4);
      asm volatile("global_load_async_to_lds_b128 %0, %1, off"
                   :: "v"(lds_off), "v"(gaddr)
                   : "memory");
    }
    asm volatile("s_wait_asynccnt 0x0" ::: "memory");
  }

  const int wave = threadIdx.x >> 5;
  const int lane = threadIdx.x & 31;
  const int task = blockIdx.x * 8 + wave;       // (b*H + h)*T + t
  const int t  = task % T_;
  const int bh = task / T_;
  const int h  = bh % H_;
  const int b  = bh / H_;

  const size_t row3c = (size_t)3 * C_;
  const float* qp = qkv + (size_t)(b * T_ + t) * row3c + h * HD_;
  const float* kb = qkv + (size_t)(b * T_) * row3c + C_ + h * HD_;
  const float* vb = qkv + (size_t)(b * T_) * row3c + 2 * C_ + h * HD_;

  s_q[wave][lane]      = qp[lane];
  s_q[wave][lane + 32] = qp[lane + 32];
  __syncthreads();

  auto score = [&](int j) -> float {
    const int off  = c_offs[j];
    const int idx  = t - off;
    const int idxc = idx < 0 ? 0 : idx;
    const v4f* kr4 = (const v4f*)(kb + (size_t)idxc * row3c);   // 256B aligned
    const v4f* q4  = (const v4f*)(&s_q[wave][0]);               // 16B aligned
    const v4f* se4 = (const v4f*)(&s_se[j * SE_PITCH]);         // 272B pitch
    float p0 = 0.f, p1 = 0.f, p2 = 0.f, p3 = 0.f;
    #pragma unroll
    for (int d4 = 0; d4 < HD_ / 4; ++d4) {
      v4f kk = kr4[d4], qq = q4[d4], ss = se4[d4];
      p0 = fmaf(qq.x * (1.0f + ss.x), kk.x, p0);
      p1 = fmaf(qq.y * (1.0f + ss.y), kk.y, p1);
      p2 = fmaf(qq.z * (1.0f + ss.z), kk.z, p2);
      p3 = fmaf(qq.w * (1.0f + ss.w), kk.w, p3);
    }
    float acc = (p0 + p1) + (p2 + p3);
    float pb = pos_bias[j * H_ + h];
    const bool distal = off > 350;          // DISTAL_THRESHOLD
    if (distal && h < 7)   pb = -10000.0f;  // local hard mask
    if (!distal && h >= 7) pb = -3.0f;      // distal soft mask
    float s = acc * 0.125f + pb;            // 1/sqrt(64)
    if (idx < 0) s = -1.0e9f;               // causal validity
    return s;
  };

  float s0 = score(lane);
  float s1 = (lane + 32 < J_) ? score(lane + 32) : -3.0e38f;

  float m = fmaxf(s0, s1);
  #pragma unroll
  for (int o = 16; o; o >>= 1) m = fmaxf(m, __shfl_xor(m, o));
  float p0 = __expf(s0 - m);
  float p1 = (lane + 32 < J_) ? __expf(s1 - m) : 0.0f;
  float sum = p0 + p1;
  #pragma unroll
  for (int o = 16; o; o >>= 1) sum += __shfl_xor(sum, o);

  s_p[wave][lane]      = p0;
  s_p[wave][lane + 32] = p1;
  __syncthreads();

  float o0 = 0.0f, o1 = 0.0f;
  #pragma unroll 4
  for (int j = 0; j < J_; ++j) {
    const float pj = s_p[wave][j];
    const int idx  = t - c_offs[j];
    const int idxc = idx < 0 ? 0 : idx;
    const float* vr = vb + (size_t)idxc * row3c;
    o0 = fmaf(pj, vr[lane],      o0);
    o1 = fmaf(pj, vr[lane + 32], o1);
  }

  const float g = if_gain[h] / sum;
  float* op = attn + (size_t)(b * T_ + t) * C_ + h * HD_;
  op[lane]      = o0 * g;
  op[lane + 32] = o1 * g;
}

// ---------------------------------------------------------------------------
extern "C" void kernel_launch(void* const* d_in, const int* in_sizes, int n_in,
                              void* d_out, int out_size, void* d_ws, size_t ws_size,
                              hipStream_t stream) {
  const float* x    = (const float*)d_in[0];   // (B,T,C)
  const float* Wqkv = (const float*)d_in[1];   // (3C,C)
  const float* Wout = (const float*)d_in[2];   // (C,C)
  const float* pb   = (const float*)d_in[3];   // (J,H)
  const float* se   = (const float*)d_in[4];   // (J,HD)
  const float* ig   = (const float*)d_in[5];   // (H)
  float* out = (float*)d_out;

  float* qkv  = (float*)d_ws;                          // (B*T, 3C) = 37.75 MB
  float* attn = qkv + (size_t)B_ * T_ * 3 * C_;        // (B*T,  C) = 12.58 MB

  const int M = B_ * T_;                               // 4096

  // 1) qkv = x @ Wqkv^T      M=4096, N=2304, K=768
  {
    const int nT = (3 * C_) / 64;                      // 36
    const int tasks = (M / 16) * nT;                   // 9216 waves
    wmma_gemm_nt<<<tasks / 8, 256, 0, stream>>>(x, Wqkv, qkv, M, 3 * C_, C_, nT);
  }
  // 2) gathered-offset attention
  {
    const int tasks = B_ * H_ * T_;                    // 49152 waves
    dsqg_attn<<<tasks / 8, 256, 0, stream>>>(qkv, pb, se, ig, attn);
  }
  // 3) out = attn @ Wout^T   M=4096, N=768, K=768
  {
    const int nT = C_ / 64;                            // 12
    const int tasks = (M / 16) * nT;                   // 3072 waves
    wmma_gemm_nt<<<tasks / 8, 256, 0, stream>>>(attn, Wout, out, M, C_, C_, nT);
  }
}